// SpikingMambaBlock_4389456577449
// MI455X (gfx1250) — compile-verified
//
#include <hip/hip_runtime.h>
#include <hip/hip_bf16.h>
#include <math.h>

#define D_MODEL   1024
#define D_STATE   16
#define D_CONV    4
#define D_INNER   2048
#define DT_RANK   64
#define BATCH     4
#define SEQ       2048
#define PROJ_OUT  4192            // 2*D_INNER + 2*D_STATE + DT_RANK
#define NROWS     (BATCH * SEQ)   // 8192

typedef __attribute__((ext_vector_type(2))) float v2f;
typedef __attribute__((ext_vector_type(8))) float v8f;

__device__ __forceinline__ float sigmoidf_(float x) { return 1.f / (1.f + __expf(-x)); }
__device__ __forceinline__ float siluf_(float x)    { return x * sigmoidf_(x); }
__device__ __forceinline__ float softplusf_(float x){ return (x > 20.f) ? x : log1pf(__expf(x)); }

// ---------------------------------------------------------------------------
// C[M,N] = A[M,K] @ B[K,N] + bias[N], optional softplus epilogue (EPI==1).
// One wave computes a 32x32 output tile with 2x2 V_WMMA_F32_16X16X4_F32
// accumulator fragments. f32 in / f32 accumulate: exact vs the f32 reference.
//
// ISA layouts (CDNA5 ISA §7.12.2):
//   A 16x4 f32 : lane l<16 -> row l,   VGPR0=K0,VGPR1=K1 ; l>=16 -> K2,K3
//   B  4x16 f32: mirrored — lane-half selects K rows {0,1} vs {2,3}, N = l%16
//   C/D 16x16  : VGPR r -> row r (lanes 0-15) / r+8 (lanes 16-31), N = l%16
// ---------------------------------------------------------------------------
template<int EPI>
__global__ __launch_bounds__(128)
void gemm_wmma_f32(const float* __restrict__ A, int lda,
                   const float* __restrict__ B, int ldb,
                   const float* __restrict__ bias,
                   float* __restrict__ C, int ldc,
                   int M, int K, int tilesN)
{
    const int wave   = blockIdx.x * (blockDim.x >> 5) + (threadIdx.x >> 5);
    const int tilesM = M >> 5;
    if (wave >= tilesM * tilesN) return;
    const int tm = wave / tilesN, tn = wave % tilesN;
    const int m0 = tm << 5,       n0 = tn << 5;
    const int lane = threadIdx.x & 31;
    const int half = lane >> 4;       // selects K pair for A/B fragments
    const int l16  = lane & 15;

    v8f acc00 = {}, acc01 = {}, acc10 = {}, acc11 = {};
    const float* Ar0 = A + (size_t)(m0 + l16)      * lda;
    const float* Ar1 = A + (size_t)(m0 + 16 + l16) * lda;

    for (int k = 0; k < K; k += 4) {
        const int ka = k + (half << 1);
        v2f a0, a1, b0, b1;
        a0.x = Ar0[ka]; a0.y = Ar0[ka + 1];
        a1.x = Ar1[ka]; a1.y = Ar1[ka + 1];
        const float* Bk0 = B + (size_t)ka * ldb + n0;   // row k + 2*half
        const float* Bk1 = Bk0 + ldb;                   // row k + 2*half + 1
        b0.x = Bk0[l16];      b0.y = Bk1[l16];
        b1.x = Bk0[16 + l16]; b1.y = Bk1[16 + l16];
        acc00 = __builtin_amdgcn_wmma_f32_16x16x4_f32(false, a0, false, b0, (short)0, acc00, false, false);
        acc01 = __builtin_amdgcn_wmma_f32_16x16x4_f32(false, a0, false, b1, (short)0, acc01, false, false);
        acc10 = __builtin_amdgcn_wmma_f32_16x16x4_f32(false, a1, false, b0, (short)0, acc10, false, false);
        acc11 = __builtin_amdgcn_wmma_f32_16x16x4_f32(false, a1, false, b1, (short)0, acc11, false, false);
    }

    const float bia0 = bias[n0 + l16];
    const float bia1 = bias[n0 + 16 + l16];
    #pragma unroll
    for (int r = 0; r < 8; ++r) {
        const int mr0 = m0 + r + (half << 3);   // C/D layout: lane-half -> +8 rows
        const int mr1 = mr0 + 16;
        float v;
        v = acc00[r] + bia0; if (EPI == 1) v = softplusf_(v); C[(size_t)mr0 * ldc + n0 + l16]      = v;
        v = acc01[r] + bia1; if (EPI == 1) v = softplusf_(v); C[(size_t)mr0 * ldc + n0 + 16 + l16] = v;
        v = acc10[r] + bia0; if (EPI == 1) v = softplusf_(v); C[(size_t)mr1 * ldc + n0 + l16]      = v;
        v = acc11[r] + bia1; if (EPI == 1) v = softplusf_(v); C[(size_t)mr1 * ldc + n0 + 16 + l16] = v;
    }
}

// ---------------------------------------------------------------------------
// Causal depthwise conv (K=4) over seq + SiLU: u[b,t,d]
// x_in lives inside proj at column offset D_INNER.
// ---------------------------------------------------------------------------
__global__ __launch_bounds__(256)
void conv_silu_kernel(const float* __restrict__ proj,
                      const float* __restrict__ cw,
                      const float* __restrict__ cb,
                      float* __restrict__ u)
{
    const size_t gid = (size_t)blockIdx.x * blockDim.x + threadIdx.x;
    if (gid >= (size_t)NROWS * D_INNER) return;
    const int d = (int)(gid % D_INNER);
    const int r = (int)(gid / D_INNER);
    const int t = r % SEQ;
    const int b = r / SEQ;
    const float w0 = cw[d * 4 + 0], w1 = cw[d * 4 + 1];
    const float w2 = cw[d * 4 + 2], w3 = cw[d * 4 + 3];
    const size_t base = ((size_t)b * SEQ) * PROJ_OUT + D_INNER + d;
    float acc = cb[d];
    if (t >= 3) acc += w0 * proj[base + (size_t)(t - 3) * PROJ_OUT];
    if (t >= 2) acc += w1 * proj[base + (size_t)(t - 2) * PROJ_OUT];
    if (t >= 1) acc += w2 * proj[base + (size_t)(t - 1) * PROJ_OUT];
    acc += w3 * proj[base + (size_t)t * PROJ_OUT];
    u[gid] = siluf_(acc);
}

// ---------------------------------------------------------------------------
// Selective-scan: one thread per (b, d) channel, h[16] in registers,
// B_ssm[b, t0:t0+CHUNK, :] staged through LDS. Only the final state is
// needed: y_last = sum_n h_L[n] * C_L[n], then gating + D skip-connection.
// ---------------------------------------------------------------------------
#define SCAN_CHUNK 256
__global__ __launch_bounds__(256)
void scan_kernel(const float* __restrict__ proj,
                 const float* __restrict__ u,
                 const float* __restrict__ dt,
                 const float* __restrict__ A_log,
                 const float* __restrict__ Dp,
                 float* __restrict__ fm)
{
    __shared__ float Bs[SCAN_CHUNK * D_STATE];   // 16 KB
    const int b = blockIdx.y;
    const int d = blockIdx.x * blockDim.x + threadIdx.x;
    const size_t baseRow = (size_t)b * SEQ;

    float Arow[D_STATE], h[D_STATE];
    #pragma unroll
    for (int n = 0; n < D_STATE; ++n) {
        Arow[n] = -__expf(A_log[d * D_STATE + n]);
        h[n] = 0.f;
    }
    const float* dtp = dt + baseRow * D_INNER + d;
    const float* up  = u  + baseRow * D_INNER + d;

    for (int t0 = 0; t0 < SEQ; t0 += SCAN_CHUNK) {
        __syncthreads();
        for (int idx = threadIdx.x; idx < SCAN_CHUNK * D_STATE; idx += blockDim.x) {
            const int tl = idx >> 4, n = idx & 15;
            Bs[idx] = proj[(baseRow + t0 + tl) * PROJ_OUT + 2 * D_INNER + n];
        }
        __syncthreads();
        for (int tl = 0; tl < SCAN_CHUNK; ++tl) {
            const size_t t = (size_t)(t0 + tl);
            const float dtv = dtp[t * D_INNER];
            const float uv  = up[t * D_INNER];
            const float du  = dtv * uv;
            const float* Bt = &Bs[tl * D_STATE];
            #pragma unroll
            for (int n = 0; n < D_STATE; ++n)
                h[n] = __expf(Arow[n] * dtv) * h[n] + du * Bt[n];
        }
    }

    const float* last = proj + (baseRow + SEQ - 1) * PROJ_OUT;
    float y = 0.f;
    #pragma unroll
    for (int n = 0; n < D_STATE; ++n)
        y += h[n] * last[2 * D_INNER + D_STATE + n];
    const float ulast = up[(size_t)(SEQ - 1) * D_INNER];
    const float z     = last[d];                    // z column of proj
    y = (y + ulast * Dp[d]) * siluf_(z);
    fm[b * D_INNER + d] = y;
}

// ---------------------------------------------------------------------------
// LayerNorm over D_INNER per batch row (4 rows).
// ---------------------------------------------------------------------------
__global__ __launch_bounds__(256)
void ln_kernel(const float* __restrict__ fm,
               const float* __restrict__ lnw,
               const float* __restrict__ lnb,
               float* __restrict__ fmn)
{
    __shared__ float red[256];
    const int b = blockIdx.x;
    const float* row = fm + b * D_INNER;

    float s = 0.f;
    for (int d = threadIdx.x; d < D_INNER; d += 256) s += row[d];
    red[threadIdx.x] = s; __syncthreads();
    for (int off = 128; off > 0; off >>= 1) {
        if (threadIdx.x < off) red[threadIdx.x] += red[threadIdx.x + off];
        __syncthreads();
    }
    const float mu = red[0] * (1.f / D_INNER);
    __syncthreads();

    float v = 0.f;
    for (int d = threadIdx.x; d < D_INNER; d += 256) {
        const float t = row[d] - mu; v += t * t;
    }
    red[threadIdx.x] = v; __syncthreads();
    for (int off = 128; off > 0; off >>= 1) {
        if (threadIdx.x < off) red[threadIdx.x] += red[threadIdx.x + off];
        __syncthreads();
    }
    const float rstd = rsqrtf(red[0] * (1.f / D_INNER) + 1e-5f);

    for (int d = threadIdx.x; d < D_INNER; d += 256)
        fmn[b * D_INNER + d] = (row[d] - mu) * rstd * lnw[d] + lnb[d];
}

// ---------------------------------------------------------------------------
// Final projection: out[4,1024] = fm_n @ out_w + out_b (tiny; one thread/out)
// ---------------------------------------------------------------------------
__global__ __launch_bounds__(256)
void outproj_kernel(const float* __restrict__ fmn,
                    const float* __restrict__ W,
                    const float* __restrict__ bias,
                    float* __restrict__ out)
{
    const int gid = blockIdx.x * blockDim.x + threadIdx.x;   // 0..4095
    const int b = gid / D_MODEL, m = gid % D_MODEL;
    float acc = bias[m];
    const float* f = fmn + b * D_INNER;
    for (int d = 0; d < D_INNER; ++d)
        acc += f[d] * W[(size_t)d * D_MODEL + m];
    out[gid] = acc;
}

// ---------------------------------------------------------------------------
extern "C" void kernel_launch(void* const* d_in, const int* in_sizes, int n_in,
                              void* d_out, int out_size, void* d_ws, size_t ws_size,
                              hipStream_t stream)
{
    const float* x      = (const float*)d_in[0];
    const float* in_w   = (const float*)d_in[1];
    const float* in_b   = (const float*)d_in[2];
    const float* conv_w = (const float*)d_in[3];
    const float* conv_b = (const float*)d_in[4];
    const float* dt_w   = (const float*)d_in[5];
    const float* dt_b   = (const float*)d_in[6];
    const float* A_log  = (const float*)d_in[7];
    const float* D_par  = (const float*)d_in[8];
    const float* out_w  = (const float*)d_in[9];
    const float* out_b  = (const float*)d_in[10];
    const float* ln_w   = (const float*)d_in[11];
    const float* ln_b   = (const float*)d_in[12];

    float* ws   = (float*)d_ws;
    float* proj = ws;                                   // NROWS * PROJ_OUT
    float* u    = proj + (size_t)NROWS * PROJ_OUT;      // NROWS * D_INNER
    float* dt   = u    + (size_t)NROWS * D_INNER;       // NROWS * D_INNER
    float* fm   = dt   + (size_t)NROWS * D_INNER;       // BATCH * D_INNER
    float* fmn  = fm   + BATCH * D_INNER;               // BATCH * D_INNER

    // 1) proj = x @ in_w + in_b                 [8192,1024] x [1024,4192]
    {
        const int tilesM = NROWS / 32, tilesN = PROJ_OUT / 32;   // 256 x 131
        const int waves  = tilesM * tilesN;
        gemm_wmma_f32<0><<<(waves + 3) / 4, 128, 0, stream>>>(
            x, D_MODEL, in_w, PROJ_OUT, in_b, proj, PROJ_OUT,
            NROWS, D_MODEL, tilesN);
    }
    // 2) u = silu(depthwise_conv(x_in) + conv_b)
    {
        const size_t total = (size_t)NROWS * D_INNER;
        conv_silu_kernel<<<(int)((total + 255) / 256), 256, 0, stream>>>(
            proj, conv_w, conv_b, u);
    }
    // 3) dt = softplus(dt_r @ dt_w + dt_b)      [8192,64] x [64,2048]
    {
        const int tilesM = NROWS / 32, tilesN = D_INNER / 32;    // 256 x 64
        const int waves  = tilesM * tilesN;
        gemm_wmma_f32<1><<<(waves + 3) / 4, 128, 0, stream>>>(
            proj + 2 * D_INNER + 2 * D_STATE, PROJ_OUT,
            dt_w, D_INNER, dt_b, dt, D_INNER,
            NROWS, DT_RANK, tilesN);
    }
    // 4) selective scan -> fm (last-timestep output, gated)
    scan_kernel<<<dim3(D_INNER / 256, BATCH), 256, 0, stream>>>(
        proj, u, dt, A_log, D_par, fm);
    // 5) layernorm
    ln_kernel<<<BATCH, 256, 0, stream>>>(fm, ln_w, ln_b, fmn);
    // 6) out = fm_n @ out_w + out_b
    outproj_kernel<<<(BATCH * D_MODEL) / 256, 256, 0, stream>>>(
        fmn, out_w, out_b, (float*)d_out);
}